// ScaledDotProductAttention_71528385347804
// MI455X (gfx1250) — compile-verified
//
#include <hip/hip_runtime.h>
#include <hip/hip_bf16.h>
#include <math.h>

typedef float v2f __attribute__((ext_vector_type(2)));
typedef float v8f __attribute__((ext_vector_type(8)));

#define B_   8
#define LQ_  512
#define LK_  512
#define D_   64
#define IT_  16        // query rows per workgroup
#define INV_TEMP 0.125f

// ---------------------------------------------------------------------------
// Main fused kernel: one block handles (batch b, 16 query rows).
//   Phase 1: s_qk = q_tile @ k^T via v_wmma_f32_16x16x4_f32 (LDS-staged)
//   Phase 2: single pass over bias (HBM-bound): online softmax, fused
//            accumulation of attn@(v+bias); raw scores spilled to attn buffer.
// ---------------------------------------------------------------------------
__global__ __launch_bounds__(256) void attn_main(
    const float* __restrict__ q, const float* __restrict__ k,
    const float* __restrict__ v, const float* __restrict__ bias,
    float* __restrict__ out,     // [B][LQ][D]
    float* __restrict__ attn,    // [B][LQ][LK]  (raw scores after this kernel)
    float* __restrict__ mrow,    // [B*LQ] running max  (workspace)
    float* __restrict__ lrow)    // [B*LQ] running sum  (workspace)
{
    __shared__ float q_s[IT_][D_];     // 4 KB
    __shared__ float skq[IT_][LK_];    // 32 KB: q@k^T scores for this tile

    const int bi  = blockIdx.y;
    const int i0  = blockIdx.x * IT_;
    const int tid = threadIdx.x;

    // ---- load q tile (16x64) into LDS, coalesced float4 ----
    {
        const int i = tid >> 4, f = tid & 15;
        *(float4*)&q_s[i][f * 4] =
            *(const float4*)(q + ((size_t)(bi * LQ_ + i0 + i) * D_ + f * 4));
    }
    __syncthreads();

    // ---- Phase 1: WMMA q@k^T. 8 waves x 4 j-tiles each covers LK=512. ----
    {
        const int wave = tid >> 5;
        const int lane = tid & 31;
        const int half = lane >> 4;   // 0: K/M lanes 0-15, 1: K+2 per ISA layout
        const int l16  = lane & 15;
        for (int t = 0; t < 4; ++t) {
            const int jt   = wave * 4 + t;
            const int jcol = jt * 16 + l16;
            const float* krow = k + (size_t)(bi * LK_ + jcol) * D_;
            v8f c = {};
            #pragma unroll
            for (int kb = 0; kb < D_; kb += 4) {
                v2f a, bb;
                // A[M=l16][K=kb+half*2+{0,1}] from q tile (LDS)
                a.x = q_s[l16][kb + half * 2 + 0];
                a.y = q_s[l16][kb + half * 2 + 1];
                // B[K][N=jcol] = k[jcol][K]  (k is L2-hot)
                bb = *(const v2f*)(krow + kb + half * 2);
                c = __builtin_amdgcn_wmma_f32_16x16x4_f32(
                        false, a, false, bb, (short)0, c, false, false);
            }
            #pragma unroll
            for (int r = 0; r < 8; ++r)
                skq[r + half * 8][jt * 16 + l16] = c[r];
        }
    }
    __syncthreads();

    // ---- Phase 2: stream bias once; online softmax + fused output ----
    const int i  = tid >> 4;      // query row within tile
    const int tg = tid & 15;      // 16-lane group position over D
    const int d0 = tg * 4;
    const float4 qv = *(const float4*)&q_s[i][d0];

    const size_t rowq = (size_t)(bi * LQ_ + i0 + i);
    const float* brow = bias + rowq * LK_ * D_;        // bias[b][i][*][*]
    const float* vrow = v + (size_t)bi * LK_ * D_;     // v[b][*][*] (L2-hot)
    float*       arow = attn + rowq * LK_;

    float4 acc = make_float4(0.f, 0.f, 0.f, 0.f);
    float m = -INFINITY, l = 0.f;

    for (int j = 0; j < LK_; j += 4) {
        // keep the HBM stream ahead (lowers to global_prefetch_b8)
        if (j + 16 < LK_)
            __builtin_prefetch(brow + (size_t)(j + 16) * D_ + d0, 0, 0);

        float4 bq[4];
        #pragma unroll
        for (int u = 0; u < 4; ++u)
            bq[u] = *(const float4*)(brow + (size_t)(j + u) * D_ + d0);

        #pragma unroll
        for (int u = 0; u < 4; ++u) {
            // partial dot q . bias over this thread's 4 d's
            float part = qv.x * bq[u].x + qv.y * bq[u].y
                       + qv.z * bq[u].z + qv.w * bq[u].w;
            // reduce across the 16-lane d-group (wave32 segments of 16)
            part += __shfl_xor(part, 1, 16);
            part += __shfl_xor(part, 2, 16);
            part += __shfl_xor(part, 4, 16);
            part += __shfl_xor(part, 8, 16);

            const float s = (skq[i][j + u] + part) * INV_TEMP;
            // spill raw score (normalized by pass 2); one lane per j
            if (((j + u) & 15) == tg) arow[j + u] = s;

            // online softmax update
            const float mn   = fmaxf(m, s);
            const float corr = __expf(m - mn);   // exp(-inf)=0 on first iter
            const float p    = __expf(s - mn);
            m = mn;
            l = l * corr + p;

            const float4 vv = *(const float4*)(vrow + (size_t)(j + u) * D_ + d0);
            acc.x = acc.x * corr + p * (vv.x + bq[u].x);
            acc.y = acc.y * corr + p * (vv.y + bq[u].y);
            acc.z = acc.z * corr + p * (vv.z + bq[u].z);
            acc.w = acc.w * corr + p * (vv.w + bq[u].w);
        }
    }

    const float invl = 1.f / l;
    *(float4*)(out + rowq * D_ + d0) =
        make_float4(acc.x * invl, acc.y * invl, acc.z * invl, acc.w * invl);
    if (tg == 0) { mrow[rowq] = m; lrow[rowq] = l; }
}

// ---------------------------------------------------------------------------
// Pass 2: normalize raw scores in place: attn = exp(s - m_row) / l_row.
// 16 MB of traffic — negligible vs the 512 MB bias stream.
// ---------------------------------------------------------------------------
__global__ __launch_bounds__(256) void attn_norm(
    float* __restrict__ attn,
    const float* __restrict__ mrow, const float* __restrict__ lrow)
{
    const size_t g = (size_t)blockIdx.x * blockDim.x + threadIdx.x; // float4 idx
    const size_t o = g * 4;
    const size_t row = o / LK_;            // LK % 4 == 0 -> row constant per f4
    const float  m   = mrow[row];
    const float  inv = 1.f / lrow[row];
    float4 s = *(float4*)(attn + o);
    s.x = __expf(s.x - m) * inv;
    s.y = __expf(s.y - m) * inv;
    s.z = __expf(s.z - m) * inv;
    s.w = __expf(s.w - m) * inv;
    *(float4*)(attn + o) = s;
}

extern "C" void kernel_launch(void* const* d_in, const int* in_sizes, int n_in,
                              void* d_out, int out_size, void* d_ws, size_t ws_size,
                              hipStream_t stream) {
    const float* q    = (const float*)d_in[0];
    const float* k    = (const float*)d_in[1];
    const float* v    = (const float*)d_in[2];
    const float* bias = (const float*)d_in[3];

    float* out  = (float*)d_out;                       // [B*LQ*D]
    float* attn = out + (size_t)B_ * LQ_ * D_;         // [B*LQ*LK]
    float* mrow = (float*)d_ws;                        // [B*LQ]
    float* lrow = mrow + (size_t)B_ * LQ_;             // [B*LQ]

    attn_main<<<dim3(LQ_ / IT_, B_), 256, 0, stream>>>(
        q, k, v, bias, out, attn, mrow, lrow);

    const size_t nquads = (size_t)B_ * LQ_ * LK_ / 4;
    attn_norm<<<dim3((unsigned)((nquads + 255) / 256)), 256, 0, stream>>>(
        attn, mrow, lrow);
}